// InnerProductAttention_91199335563800
// MI455X (gfx1250) — compile-verified
//
#include <hip/hip_runtime.h>
#include <math.h>

// ---------------------------------------------------------------------------
// Quantum-kernel attention:
//   out[b,sq,sk] = |<0000| U_circuit |psi(b,sq,sk)>|^2
// psi is a product state from angles half_q = 0.5*pi*(tanh(q_q)-tanh(k_q)).
// a0 = sum_j w_j * psi_j,  w = U^T e0 (16 complex consts from `weights`),
// psi_j = p01[i]*p23[k], j = 4i+k  (p01/p23 = cos/sin products).
// The 16-term dot product is done as D = Psi(16x16) x W(16x16) via four
// chained V_WMMA_F32_16X16X4_F32, W cols 0/1 = w_re/w_im.
// ---------------------------------------------------------------------------

typedef float v2f __attribute__((ext_vector_type(2)));
typedef float v8f __attribute__((ext_vector_type(8)));

struct c32 { float x, y; };
__device__ inline c32 cmul(c32 a, c32 b) { return {a.x*b.x - a.y*b.y, a.x*b.y + a.y*b.x}; }
__device__ inline c32 cadd(c32 a, c32 b) { return {a.x + b.x, a.y + b.y}; }

// --- Kernel 1: compute w = U_total^T e0 (row 0 of circuit unitary). --------
// Forward circuit per layer: R(l,0)..R(l,3) then CNOT(0,1),CNOT(1,2),CNOT(2,3).
// Transposed-reverse: layer 1 then 0; within layer: CNOT(2,3),CNOT(1,2),
// CNOT(0,1) (CNOT^T==CNOT), then R(l,3)^T .. R(l,0)^T.
// Qubit q -> bit (3-q) of basis index j (qubit0 is MSB of abcd).
__global__ void circuit_row_kernel(const float* __restrict__ w, float* __restrict__ wbuf) {
    if (threadIdx.x != 0 || blockIdx.x != 0) return;
    c32 r[16];
    for (int j = 0; j < 16; ++j) r[j] = {0.f, 0.f};
    r[0] = {1.f, 0.f};
    for (int l = 1; l >= 0; --l) {
        // CNOTs (transposed == same), reverse order q = 2,1,0
        for (int q = 2; q >= 0; --q) {
            int cb = 1 << (3 - q);        // control bit
            int tb = 1 << (2 - q);        // target bit
            for (int j = 0; j < 16; ++j) {
                if ((j & cb) && !(j & tb)) {
                    c32 tmp = r[j]; r[j] = r[j | tb]; r[j | tb] = tmp;
                }
            }
        }
        // 1-qubit gates transposed, reverse order q = 3..0
        for (int q = 3; q >= 0; --q) {
            float phi = w[(l * 4 + q) * 3 + 0];
            float th  = w[(l * 4 + q) * 3 + 1];
            float om  = w[(l * 4 + q) * 3 + 2];
            float ct = cosf(0.5f * th), st = sinf(0.5f * th);
            float ap = 0.5f * (phi + om), am = 0.5f * (phi - om);
            c32 ep = { cosf(ap), -sinf(ap) };   // exp(-i*(phi+om)/2)
            c32 em = { cosf(am),  sinf(am) };   // exp(+i*(phi-om)/2)
            c32 U00 = {  ep.x * ct,  ep.y * ct };
            c32 U01 = { -em.x * st, -em.y * st };
            c32 U10 = {  em.x * st, -em.y * st };  // conj(em)*st
            c32 U11 = {  ep.x * ct, -ep.y * ct };  // conj(ep)*ct
            int qb = 1 << (3 - q);
            for (int j = 0; j < 16; ++j) {
                if (j & qb) continue;
                c32 r0 = r[j], r1 = r[j | qb];
                // transposed gate: n0 = U00*r0 + U10*r1 ; n1 = U01*r0 + U11*r1
                c32 n0 = cadd(cmul(U00, r0), cmul(U10, r1));
                c32 n1 = cadd(cmul(U01, r0), cmul(U11, r1));
                r[j] = n0; r[j | qb] = n1;
            }
        }
    }
    for (int j = 0; j < 16; ++j) { wbuf[j] = r[j].x; wbuf[16 + j] = r[j].y; }
}

// --- Kernel 2: tanh tables. thq[b*512+s][0..3], thk likewise. --------------
__global__ void tanh_kernel(const float* __restrict__ q, const float* __restrict__ k,
                            float* __restrict__ thq, float* __restrict__ thk) {
    int row = blockIdx.x * blockDim.x + threadIdx.x;    // 0..4095
    const float* src; float* dst; int r;
    if (row < 2048) { src = q; dst = thq; r = row; }
    else            { src = k; dst = thk; r = row - 2048; }
    float4 v = *(const float4*)(src + (size_t)r * 64);
    float4 o;
    o.x = tanhf(v.x); o.y = tanhf(v.y); o.z = tanhf(v.z); o.w = tanhf(v.w);
    *(float4*)(dst + (size_t)r * 4) = o;
}

// --- Kernel 3: main WMMA kernel. 16 outputs (one sk-tile) per wave-iter. ---
#define TILES_PER_WAVE 8

__global__ __launch_bounds__(256) void qattn_kernel(const float* __restrict__ thq,
                                                    const float* __restrict__ thk,
                                                    const float* __restrict__ wbuf,
                                                    float* __restrict__ out) {
    const int lane = threadIdx.x & 31;
    const int wave = blockIdx.x * (blockDim.x >> 5) + (threadIdx.x >> 5);
    const int col  = lane & 15;
    const bool hi  = lane >= 16;

    // B operand (constant across tiles): B[t].x = Wmat[4t+2*hi][col],
    // B[t].y = Wmat[4t+2*hi+1][col]; Wmat[j][0]=w_re[j], Wmat[j][1]=w_im[j].
    v2f B[4];
#pragma unroll
    for (int t = 0; t < 4; ++t) {
        int j0 = 4 * t + (hi ? 2 : 0);
        B[t].x = (col < 2) ? wbuf[col * 16 + j0]     : 0.f;
        B[t].y = (col < 2) ? wbuf[col * 16 + j0 + 1] : 0.f;
    }

    const float HPI = 1.57079632679489661923f;
    const int tile0 = wave * TILES_PER_WAVE;
#pragma unroll 1
    for (int it = 0; it < TILES_PER_WAVE; ++it) {
        int T   = tile0 + it;           // 0..65535
        int bq  = T >> 5;               // b*512 + sq
        int sk0 = (T & 31) << 4;
        int b   = bq >> 9;
        int sk  = sk0 + col;            // this lane's element

        float4 hq = *(const float4*)(thq + (size_t)bq * 4);
        float4 hk = *(const float4*)(thk + ((size_t)(b * 512 + sk)) * 4);

        float s0, c0, s1, c1, s2, c2, s3, c3;
        __sincosf((hq.x - hk.x) * HPI, &s0, &c0);
        __sincosf((hq.y - hk.y) * HPI, &s1, &c1);
        __sincosf((hq.z - hk.z) * HPI, &s2, &c2);
        __sincosf((hq.w - hk.w) * HPI, &s3, &c3);

        float p01_0 = c0 * c1, p01_1 = c0 * s1, p01_2 = s0 * c1, p01_3 = s0 * s1;
        float u2 = hi ? s2 : c2;              // lo lanes: k=0,1  hi lanes: k=2,3
        float pA = u2 * c3, pB = u2 * s3;

        v8f acc = {0.f, 0.f, 0.f, 0.f, 0.f, 0.f, 0.f, 0.f};
        v2f A;
        A.x = p01_0 * pA; A.y = p01_0 * pB;
        acc = __builtin_amdgcn_wmma_f32_16x16x4_f32(false, A, false, B[0], (short)0, acc, false, false);
        A.x = p01_1 * pA; A.y = p01_1 * pB;
        acc = __builtin_amdgcn_wmma_f32_16x16x4_f32(false, A, false, B[1], (short)0, acc, false, false);
        A.x = p01_2 * pA; A.y = p01_2 * pB;
        acc = __builtin_amdgcn_wmma_f32_16x16x4_f32(false, A, false, B[2], (short)0, acc, false, false);
        A.x = p01_3 * pA; A.y = p01_3 * pB;
        acc = __builtin_amdgcn_wmma_f32_16x16x4_f32(false, A, false, B[3], (short)0, acc, false, false);

        // D[m][0]=re (lanes 0/16), D[m][1]=im (lanes 1/17); pair via xor-1.
        float sq[8];
#pragma unroll
        for (int rI = 0; rI < 8; ++rI) {
            float mine  = acc[rI];
            float other = __shfl_xor(mine, 1, 32);
            sq[rI] = mine * mine + other * other;
        }
        if (col == 0) {
            int mbase = hi ? 8 : 0;
            float* dst = out + (size_t)bq * 512 + sk0 + mbase;
            float4 o0 = { sq[0], sq[1], sq[2], sq[3] };
            float4 o1 = { sq[4], sq[5], sq[6], sq[7] };
            *(float4*)(dst)     = o0;
            *(float4*)(dst + 4) = o1;
        }
    }
}

extern "C" void kernel_launch(void* const* d_in, const int* in_sizes, int n_in,
                              void* d_out, int out_size, void* d_ws, size_t ws_size,
                              hipStream_t stream) {
    const float* query = (const float*)d_in[0];   // (4,512,64)
    const float* key   = (const float*)d_in[1];   // (4,512,64)
    const float* wts   = (const float*)d_in[2];   // (2,4,3)
    float* out = (float*)d_out;                   // (4,512,512)
    float* ws  = (float*)d_ws;

    float* wbuf = ws;               // 32 floats: w_re[16], w_im[16]
    float* thq  = ws + 64;          // 4*512*4 floats
    float* thk  = ws + 64 + 8192;   // 4*512*4 floats

    circuit_row_kernel<<<1, 32, 0, stream>>>(wts, wbuf);
    tanh_kernel<<<16, 256, 0, stream>>>(query, key, thq, thk);

    // 4*512*(512/16) = 65536 tiles; 8 tiles/wave, 8 waves/block -> 1024 blocks
    qattn_kernel<<<1024, 256, 0, stream>>>(thq, thk, wbuf, out);
}